// KKTLossGraph_86904368268048
// MI455X (gfx1250) — compile-verified
//
#include <hip/hip_runtime.h>
#include <hip/hip_bf16.h>
#include <stdint.h>

// ---------------------------------------------------------------------------
// KKT loss graph for MI455X (gfx1250, wave32).
// Memory-bound: ~100MB edge stream @ 23.3 TB/s; x/lam/Ax/ATlam are L2-resident.
// CDNA5 paths used: async global->LDS edge staging (ASYNCcnt), native f32
// global atomics, V_WMMA_F32_16X16X4_F32 ones-matrix wave reductions.
// ---------------------------------------------------------------------------

#define EDGE_BLOCK 256
#define TILE 1024  // edges per block per stage (4 per thread, 16B async b128 each)

typedef float v2f __attribute__((ext_vector_type(2)));
typedef float v8f __attribute__((ext_vector_type(8)));

#if defined(__gfx1250__) &&                                            \
    __has_builtin(__builtin_amdgcn_global_load_async_to_lds_b128) &&   \
    __has_builtin(__builtin_amdgcn_s_wait_asynccnt)
#define USE_ASYNC_LDS 1
#else
#define USE_ASYNC_LDS 0
#endif

#if USE_ASYNC_LDS
// The builtin takes (AS1 int4* gaddr, AS3 int4* ldsaddr, imm offset, imm cpol).
// Cast through uintptr_t: AS1 keeps the full VA; AS3 keeps the low 32 bits,
// which per the ISA generic->LDS mapping is the LDS
// offset.
typedef int async_b128_t __attribute__((vector_size(4 * sizeof(int))));
#define TO_GLOBAL(p) \
  ((__attribute__((address_space(1))) async_b128_t*)(uintptr_t)(p))
#define TO_LDS(p) \
  ((__attribute__((address_space(3))) async_b128_t*)(uint32_t)(uintptr_t)(p))
#endif

// --------------------------- zero workspace --------------------------------
__global__ void kkt_zero_kernel(float* __restrict__ p, int n) {
  int i = blockIdx.x * blockDim.x + threadIdx.x;
  const int stride = gridDim.x * blockDim.x;
  for (; i < n; i += stride) p[i] = 0.0f;
}

// --------------------------- edge scatter ----------------------------------
// Ax[rows[e]]    += attr[e] * x[cols[e]]
// ATlam[cols[e]] += attr[e] * lam[rows[e]]
__global__ __launch_bounds__(EDGE_BLOCK) void kkt_edge_kernel(
    const int* __restrict__ rows, const int* __restrict__ cols,
    const float* __restrict__ attr, const float* __restrict__ x,
    const float* __restrict__ lam, float* __restrict__ Ax,
    float* __restrict__ ATlam, int E) {
  __shared__ int s_rows[TILE];
  __shared__ int s_cols[TILE];
  __shared__ float s_attr[TILE];
  const int t = threadIdx.x;
  const int numTiles = E / TILE;

  for (int tile = blockIdx.x; tile < numTiles; tile += gridDim.x) {
    const int base = tile * TILE;
#if USE_ASYNC_LDS
    // Each lane copies 16B (4 elements) of each array: 3 async b128 per lane.
    __builtin_amdgcn_global_load_async_to_lds_b128(
        TO_GLOBAL(rows + base + 4 * t), TO_LDS(&s_rows[4 * t]), 0, 0);
    __builtin_amdgcn_global_load_async_to_lds_b128(
        TO_GLOBAL(cols + base + 4 * t), TO_LDS(&s_cols[4 * t]), 0, 0);
    __builtin_amdgcn_global_load_async_to_lds_b128(
        TO_GLOBAL(attr + base + 4 * t), TO_LDS(&s_attr[4 * t]), 0, 0);
    __builtin_amdgcn_s_wait_asynccnt(0);
    __syncthreads();
#else
#pragma unroll
    for (int k = 0; k < TILE / EDGE_BLOCK; ++k) {
      s_rows[t + EDGE_BLOCK * k] = rows[base + t + EDGE_BLOCK * k];
      s_cols[t + EDGE_BLOCK * k] = cols[base + t + EDGE_BLOCK * k];
      s_attr[t + EDGE_BLOCK * k] = attr[base + t + EDGE_BLOCK * k];
    }
    __syncthreads();
#endif
#pragma unroll
    for (int k = 0; k < TILE / EDGE_BLOCK; ++k) {
      const int e = t + EDGE_BLOCK * k;  // bank-conflict-free LDS reads
      const int r = s_rows[e];
      const int ci = s_cols[e];
      const float a = s_attr[e];
      unsafeAtomicAdd(&Ax[r], a * x[ci]);      // global_atomic_add_f32 (L2)
      unsafeAtomicAdd(&ATlam[ci], a * lam[r]);
    }
    __syncthreads();
  }
  // Remainder edges (E not multiple of TILE)
  for (int i = numTiles * TILE + blockIdx.x * EDGE_BLOCK + t; i < E;
       i += gridDim.x * EDGE_BLOCK) {
    const int r = rows[i];
    const int ci = cols[i];
    const float a = attr[i];
    unsafeAtomicAdd(&Ax[r], a * x[ci]);
    unsafeAtomicAdd(&ATlam[ci], a * lam[r]);
  }
}

// --------------------- WMMA wave-sum (32 lanes -> scalar) ------------------
// A(16x4) = per-lane partials (any placement is fine for a total sum),
// B(4x16) = ones -> D[m][n] = rowsum_m. Each lane's 8 D regs hold the rowsums
// of its half (rows 0-7 for lanes<16, rows 8-15 for lanes>=16); one
// shfl_xor(16) combines the halves. Exact f32 math, real v_wmma work.
__device__ __forceinline__ float wave_sum(float p) {
#if defined(__gfx1250__) && __has_builtin(__builtin_amdgcn_wmma_f32_16x16x4_f32)
  v2f a;
  a.x = p;
  a.y = 0.0f;
  v2f ones;
  ones.x = 1.0f;
  ones.y = 1.0f;
  v8f c = {};
  v8f d = __builtin_amdgcn_wmma_f32_16x16x4_f32(false, a, false, ones,
                                                (short)0, c, false, false);
  float s = 0.0f;
#pragma unroll
  for (int i = 0; i < 8; ++i) s += d[i];
  s += __shfl_xor(s, 16, 32);
  return s;
#else
  float s = p;
#pragma unroll
  for (int off = 16; off > 0; off >>= 1) s += __shfl_xor(s, off, 32);
  return s;
#endif
}

// --------------------------- residual reduce -------------------------------
__global__ __launch_bounds__(256) void kkt_reduce_kernel(
    const float* __restrict__ Ax, const float* __restrict__ ATlam,
    const float* __restrict__ lam, const float* __restrict__ b,
    const float* __restrict__ c, float* __restrict__ accum, int sum_m,
    int sum_n) {
  float pr = 0.0f, du = 0.0f, cs = 0.0f, st = 0.0f;
  const int tid = blockIdx.x * blockDim.x + threadIdx.x;
  const int stride = gridDim.x * blockDim.x;
  for (int i = tid; i < sum_m; i += stride) {
    const float axmb = Ax[i] - b[i];
    const float l = lam[i];
    const float p = fmaxf(axmb, 0.0f);  // relu(Ax-b)
    const float d = fmaxf(-l, 0.0f);    // relu(-lam)
    const float comp = l * axmb;        // complementarity
    pr = fmaf(p, p, pr);
    du = fmaf(d, d, du);
    cs = fmaf(comp, comp, cs);
  }
  for (int i = tid; i < sum_n; i += stride) {
    const float s = ATlam[i] + c[i];  // stationarity
    st = fmaf(s, s, st);
  }
  // Uniform control flow here: EXEC is all-ones for WMMA.
  pr = wave_sum(pr);
  du = wave_sum(du);
  st = wave_sum(st);
  cs = wave_sum(cs);
  if ((threadIdx.x & 31) == 0) {
    unsafeAtomicAdd(&accum[0], pr);
    unsafeAtomicAdd(&accum[1], du);
    unsafeAtomicAdd(&accum[2], st);
    unsafeAtomicAdd(&accum[3], cs);
  }
}

// ------------------------------ finalize -----------------------------------
__global__ void kkt_finalize_kernel(const float* __restrict__ accum,
                                    const int* __restrict__ Bp,
                                    const int* __restrict__ mp,
                                    const int* __restrict__ np,
                                    float* __restrict__ out) {
  if (blockIdx.x == 0 && threadIdx.x == 0) {
    const float m = (float)*mp;
    const float n = (float)*np;
    const float B = (float)*Bp;
    out[0] = (0.1f * accum[0] / m + 0.1f * accum[1] / m + 0.6f * accum[2] / n +
              0.2f * accum[3] / m) /
             B;
  }
}

// ---------------------------------------------------------------------------
extern "C" void kernel_launch(void* const* d_in, const int* in_sizes, int n_in,
                              void* d_out, int out_size, void* d_ws,
                              size_t ws_size, hipStream_t stream) {
  const float* x = (const float*)d_in[0];     // [sum_n]
  const float* lam = (const float*)d_in[1];   // [sum_m]
  const int* rows = (const int*)d_in[2];      // [E]
  const int* cols = (const int*)d_in[3];      // [E]
  const float* attr = (const float*)d_in[4];  // [E]
  const float* b = (const float*)d_in[5];     // [sum_m]
  const float* c = (const float*)d_in[6];     // [sum_n]
  const int* Bp = (const int*)d_in[7];        // scalar (device)
  const int* mp = (const int*)d_in[8];        // scalar (device)
  const int* np = (const int*)d_in[9];        // scalar (device)

  const int sum_n = in_sizes[0];
  const int sum_m = in_sizes[1];
  const int E = in_sizes[2];

  float* Ax = (float*)d_ws;      // [sum_m]
  float* ATlam = Ax + sum_m;     // [sum_n]
  float* accum = ATlam + sum_n;  // [4]

  // 1) zero Ax/ATlam/accum (workspace is poisoned, must re-zero every call)
  const int ztotal = sum_m + sum_n + 4;
  int zgrid = (ztotal + 255) / 256;
  if (zgrid > 1024) zgrid = 1024;
  kkt_zero_kernel<<<zgrid, 256, 0, stream>>>(Ax, ztotal);

  // 2) edge scatter
  int numTiles = E / TILE;
  int egrid = numTiles > 0 ? numTiles : 1;
  if (egrid > 2048) egrid = 2048;
  kkt_edge_kernel<<<egrid, EDGE_BLOCK, 0, stream>>>(rows, cols, attr, x, lam,
                                                    Ax, ATlam, E);

  // 3) residual reduce
  kkt_reduce_kernel<<<1024, 256, 0, stream>>>(Ax, ATlam, lam, b, c, accum,
                                              sum_m, sum_n);

  // 4) finalize to scalar
  kkt_finalize_kernel<<<1, 32, 0, stream>>>(accum, Bp, mp, np, (float*)d_out);
}